// ActorCriticNet_56882546868897
// MI455X (gfx1250) — compile-verified
//
#include <hip/hip_runtime.h>
#include <hip/hip_bf16.h>

// ---------------------------------------------------------------------------
// GCN 3-layer forward + heads for MI455X (gfx1250, wave32).
// Bottleneck = edge scatter-add (~1.6 GB of L2-resident traffic); GEMMs are
// tiny (0.82 GFLOP/layer) so we keep full f32 precision and use
// V_WMMA_F32_16X16X4_F32 for the dense layers.
// ---------------------------------------------------------------------------

typedef __attribute__((ext_vector_type(2))) float v2f;
typedef __attribute__((ext_vector_type(8))) float v8f;

#define HID 64

// ---------------- zero buffers ----------------
__global__ void zero_f4_kernel(float4* __restrict__ p, int n4) {
    int i = blockIdx.x * blockDim.x + threadIdx.x;
    if (i < n4) p[i] = make_float4(0.f, 0.f, 0.f, 0.f);
}

__global__ void zero_f_kernel(float* __restrict__ p, int n) {
    int i = blockIdx.x * blockDim.x + threadIdx.x;
    if (i < n) p[i] = 0.f;
}

// ---------------- edge scatter-add: agg[dst] += h[src] ----------------
// One wave (32 lanes) per edge; each lane moves a contiguous float2 of the
// 64-float row => coalesced 256B gather + contiguous f32 atomics.
__global__ __launch_bounds__(256) void scatter_add_kernel(
    const float* __restrict__ h, const int* __restrict__ src,
    const int* __restrict__ dst, float* __restrict__ agg, int n_edges) {
    int gt = blockIdx.x * blockDim.x + threadIdx.x;
    int e = gt >> 5;
    if (e >= n_edges) return;
    int lane = gt & 31;

    if (e + 1024 < n_edges) {           // speculative stream prefetch (global_prefetch_b8)
        __builtin_prefetch(&src[e + 1024], 0, 1);
        __builtin_prefetch(&dst[e + 1024], 0, 1);
    }

    int s = src[e];
    int d = dst[e];
    const float2 v = *(const float2*)(h + (size_t)s * HID + lane * 2);
    float* ap = agg + (size_t)d * HID + lane * 2;
    atomicAdd(ap,     v.x);
    atomicAdd(ap + 1, v.y);
}

// ---------------- dense layer: out = relu(A @ W + b) via WMMA f32 ----------
// A: [n_tiles*16, 64], W: [64, 64] row-major (fan_in, fan_out), b: [64].
// 8 waves/block, one 16-row M-tile per wave, W+bias staged in LDS.
__global__ __launch_bounds__(256) void gemm_bias_relu_kernel(
    const float* __restrict__ A, const float* __restrict__ W,
    const float* __restrict__ bias, float* __restrict__ out, int n_tiles) {
    __shared__ float Wl[HID * HID];
    __shared__ float bl[HID];

    // cooperative LDS fill of the 16KB weight matrix
    {
        const float4* W4 = (const float4*)W;
        float4* Wl4 = (float4*)Wl;
        for (int i = threadIdx.x; i < HID * HID / 4; i += 256) Wl4[i] = W4[i];
        if (threadIdx.x < HID) bl[threadIdx.x] = bias[threadIdx.x];
    }
    __syncthreads();

    const int wave = threadIdx.x >> 5;
    const int lane = threadIdx.x & 31;
    const int tile = blockIdx.x * 8 + wave;
    if (tile >= n_tiles) return;            // wave-uniform: EXEC stays all-ones

    const int m0    = tile * 16;
    const int mrow  = lane & 15;
    const int khalf = lane >> 4;            // 0: K+0/K+1 lanes, 1: K+2/K+3 lanes
    const float* arow = A + (size_t)(m0 + mrow) * HID + 2 * khalf;

    v8f acc[4] = {};                        // 4 n-blocks of 16 columns

#pragma unroll
    for (int k = 0; k < HID; k += 4) {
        // A fragment 16x4 (ISA 7.12.2 layout)
        const float2 av = *(const float2*)(arow + k);
        v2f a; a.x = av.x; a.y = av.y;
        const int kb = k + 2 * khalf;
#pragma unroll
        for (int j = 0; j < 4; ++j) {
            const int n = (lane & 15) + 16 * j;
            v2f bf;
            bf.x = Wl[kb * HID + n];
            bf.y = Wl[(kb + 1) * HID + n];
            acc[j] = __builtin_amdgcn_wmma_f32_16x16x4_f32(
                false, a, false, bf, (short)0, acc[j], false, false);
        }
    }

    // epilogue: +bias, ReLU, store. VGPR r of C/D holds row r (lanes 0-15)
    // or row 8+r (lanes 16-31), column = (lane&15) + 16*j.
#pragma unroll
    for (int j = 0; j < 4; ++j) {
        const int n = (lane & 15) + 16 * j;
        const float bn = bl[n];
        const int rbase = m0 + 8 * khalf;
#pragma unroll
        for (int r = 0; r < 8; ++r) {
            float val = acc[j][r] + bn;
            val = fmaxf(val, 0.f);
            out[(size_t)(rbase + r) * HID + n] = val;
        }
    }
}

// ---------------- column sum for mean_nodes ----------------
__global__ __launch_bounds__(256) void colsum_kernel(
    const float* __restrict__ h, float* __restrict__ sum64, int n_nodes) {
    const int col = threadIdx.x & 63;
    const int rowsPerBlk = 256 / 64;
    int row = blockIdx.x * rowsPerBlk + (threadIdx.x >> 6);
    const int stride = gridDim.x * rowsPerBlk;
    float acc = 0.f;
    for (; row < n_nodes; row += stride) acc += h[(size_t)row * HID + col];
    atomicAdd(&sum64[col], acc);
}

// ---------------- pi_nodes = h @ Wpg + bpg ----------------
// one wave per node; wave-shuffle reduction
__global__ __launch_bounds__(256) void pi_nodes_kernel(
    const float* __restrict__ h, const float* __restrict__ Wpg,
    const float* __restrict__ bpg, float* __restrict__ out, int n_nodes) {
    int gt = blockIdx.x * blockDim.x + threadIdx.x;
    int node = gt >> 5;
    if (node >= n_nodes) return;
    int lane = gt & 31;
    const float2 hv = *(const float2*)(h + (size_t)node * HID + lane * 2);
    const float2 wv = *(const float2*)(Wpg + lane * 2);
    float p = hv.x * wv.x + hv.y * wv.y;
#pragma unroll
    for (int off = 16; off > 0; off >>= 1) p += __shfl_xor(p, off, 32);
    if (lane == 0) out[node] = p + bpg[0];
}

// ---------------- heads: pi_done, v from mean node vector ----------------
__global__ void heads_kernel(const float* __restrict__ sum64,
                             const float* __restrict__ Wpd, const float* __restrict__ bpd,
                             const float* __restrict__ Wv,  const float* __restrict__ bv,
                             float* __restrict__ out, float invN, int n_nodes) {
    int lane = threadIdx.x;  // 32 threads
    float m0 = sum64[lane * 2]     * invN;
    float m1 = sum64[lane * 2 + 1] * invN;
    float pd = m0 * Wpd[lane * 2] + m1 * Wpd[lane * 2 + 1];
    float vv = m0 * Wv[lane * 2]  + m1 * Wv[lane * 2 + 1];
#pragma unroll
    for (int off = 16; off > 0; off >>= 1) {
        pd += __shfl_xor(pd, off, 32);
        vv += __shfl_xor(vv, off, 32);
    }
    if (lane == 0) {
        out[n_nodes]     = pd + bpd[0];   // pi_done
        out[n_nodes + 1] = vv + bv[0];    // v
    }
}

// ---------------------------------------------------------------------------
extern "C" void kernel_launch(void* const* d_in, const int* in_sizes, int n_in,
                              void* d_out, int out_size, void* d_ws, size_t ws_size,
                              hipStream_t stream) {
    const float* x   = (const float*)d_in[0];
    const int*   src = (const int*)  d_in[1];
    const int*   dst = (const int*)  d_in[2];
    const float* W1  = (const float*)d_in[3];
    const float* b1  = (const float*)d_in[4];
    const float* W2  = (const float*)d_in[5];
    const float* b2  = (const float*)d_in[6];
    const float* W3  = (const float*)d_in[7];
    const float* b3  = (const float*)d_in[8];
    const float* Wpg = (const float*)d_in[9];
    const float* bpg = (const float*)d_in[10];
    const float* Wpd = (const float*)d_in[11];
    const float* bpd = (const float*)d_in[12];
    const float* Wv  = (const float*)d_in[13];
    const float* bv  = (const float*)d_in[14];

    const int n_nodes = in_sizes[0] / HID;   // 100000
    const int n_edges = in_sizes[1];         // 1000000
    float* out = (float*)d_out;

    // workspace layout
    float* agg   = (float*)d_ws;                       // [n_nodes * 64]
    float* h     = agg + (size_t)n_nodes * HID;        // [n_nodes * 64]
    float* sum64 = h   + (size_t)n_nodes * HID;        // [64]

    const int nelem   = n_nodes * HID;
    const int zeroBlk = (nelem / 4 + 255) / 256;
    const int scatBlk = (int)(((size_t)n_edges * 32 + 255) / 256);
    const int n_tiles = (n_nodes + 15) / 16;           // 6250 (exact)
    const int gemmBlk = (n_tiles + 7) / 8;

    // ---- layer 1: x -> agg -> h ----
    zero_f4_kernel<<<zeroBlk, 256, 0, stream>>>((float4*)agg, nelem / 4);
    scatter_add_kernel<<<scatBlk, 256, 0, stream>>>(x, src, dst, agg, n_edges);
    gemm_bias_relu_kernel<<<gemmBlk, 256, 0, stream>>>(agg, W1, b1, h, n_tiles);

    // ---- layer 2: h -> agg -> h ----
    zero_f4_kernel<<<zeroBlk, 256, 0, stream>>>((float4*)agg, nelem / 4);
    scatter_add_kernel<<<scatBlk, 256, 0, stream>>>(h, src, dst, agg, n_edges);
    gemm_bias_relu_kernel<<<gemmBlk, 256, 0, stream>>>(agg, W2, b2, h, n_tiles);

    // ---- layer 3: h -> agg -> h ----
    zero_f4_kernel<<<zeroBlk, 256, 0, stream>>>((float4*)agg, nelem / 4);
    scatter_add_kernel<<<scatBlk, 256, 0, stream>>>(h, src, dst, agg, n_edges);
    gemm_bias_relu_kernel<<<gemmBlk, 256, 0, stream>>>(agg, W3, b3, h, n_tiles);

    // ---- heads ----
    zero_f_kernel<<<1, 64, 0, stream>>>(sum64, HID);
    colsum_kernel<<<256, 256, 0, stream>>>(h, sum64, n_nodes);

    const int piBlk = (int)(((size_t)n_nodes * 32 + 255) / 256);
    pi_nodes_kernel<<<piBlk, 256, 0, stream>>>(h, Wpg, bpg, out, n_nodes);
    heads_kernel<<<1, 32, 0, stream>>>(sum64, Wpd, bpd, Wv, bv, out,
                                       1.0f / (float)n_nodes, n_nodes);
}